// Net_31671088841317
// MI455X (gfx1250) — compile-verified
//
#include <hip/hip_runtime.h>

typedef __attribute__((ext_vector_type(2))) float v2f;
typedef __attribute__((ext_vector_type(8))) float v8f;
typedef __attribute__((ext_vector_type(2))) unsigned int v2u;

constexpr int NB  = 256;          // batch
constexpr int NT  = 8192;         // timesteps
constexpr int NH  = 16;           // hidden
constexpr int CH  = 16;           // timesteps per LDS chunk
constexpr int NCH = NT / CH;      // 512 chunks
constexpr int RSTR = 20;          // hist batch-row stride (floats, 16B aligned)
constexpr int SLOT = NH * RSTR + 4; // 324 floats/step: slot stride ≡ 4 mod 64 banks

#if __has_builtin(__builtin_amdgcn_tanhf)
__device__ __forceinline__ float fast_tanh(float v) { return __builtin_amdgcn_tanhf(v); }
#else
__device__ __forceinline__ float fast_tanh(float v) { return tanhf(v); }
#endif

#if __has_builtin(__builtin_amdgcn_permlane16_swap)
#define HAVE_PLSWAP 1
#else
#define HAVE_PLSWAP 0
#endif

// Exchange lane-halves between two registers:
//   on exit: x = {x[0:15], y[0:15]}, y = {x[16:31], y[16:31]}
__device__ __forceinline__ void half_swap_pair(float& x, float& y, int hi) {
#if HAVE_PLSWAP
    v2u r = __builtin_amdgcn_permlane16_swap(__float_as_uint(x), __float_as_uint(y),
                                             false, false);
    x = __uint_as_float(r[0]);
    y = __uint_as_float(r[1]);
#else
    float sx = __uint_as_float((unsigned)__builtin_amdgcn_ds_swizzle(
                   (int)__float_as_uint(x), 0x401F));   // SWAPX16
    float sy = __uint_as_float((unsigned)__builtin_amdgcn_ds_swizzle(
                   (int)__float_as_uint(y), 0x401F));
    float nx = hi ? sy : x;
    float ny = hi ? y  : sx;
    x = nx; y = ny;
#endif
}

// Transposed recurrence, state G = h^T kept as the WMMA *B* operand:
//   G' = tanh([W_ih | bias]·[x_t ; 1] + W_hh·G)
// The input projection rides as a 5th K=4 WMMA chunk whose operands do not
// depend on the recurrent state (issues ahead of the serial chain).
__global__ __launch_bounds__(64) void rnn_fused_wmma(
    const float* __restrict__ x,     const float* __restrict__ h0,
    const float* __restrict__ W_ih,  const float* __restrict__ W_hh,
    const float* __restrict__ b_ih,  const float* __restrict__ b_hh,
    const float* __restrict__ W_out, const float* __restrict__ b_out,
    float* __restrict__ out)
{
    __shared__ float xbuf[2][NH * CH];     // staged x, [batch-row][tt]   (2 KB)
    __shared__ float hist[2][CH * SLOT];   // G history, batch-major rows (~41 KB)

    const int lane = threadIdx.x & 31;
    const int wave = threadIdx.x >> 5;
    const int lo   = lane & 15;            // batch col (B/C/D) / W row (A)
    const int hi   = lane >> 4;            // lane-half selector
    const int b0   = blockIdx.x * 16;      // batch tile base

    if (wave == 1) {
        // prologue: stage x chunk 0 (rows 64B-aligned, float4 loads)
        #pragma unroll
        for (int i = 0; i < 2; ++i) {
            const int idx = lane + 32 * i;
            const int m = idx >> 2, q = idx & 3;
            const float4 v = ((const float4*)(x + (size_t)(b0 + m) * NT))[q];
            ((float4*)(&xbuf[0][m * CH]))[q] = v;
        }
    }
    __syncthreads();

    if (wave == 0) {
        // ---- persistent per-lane state ----
        // A5 = [W_ih | bias | 0 | 0]: a5[v] = A5[m=lo, k=v+2hi]
        v2f a5;
        {
            v2f t = { hi ? 0.0f : W_ih[lo],
                      hi ? 0.0f : (b_ih[lo] + b_hh[lo]) };
            a5 = t;
        }
        const float oneMask = hi ? 0.0f : 1.0f;   // B5 row k=1
        // A-layout W chunks: wm[kk][v] = W_hh[m=lo, k=4kk+v+2hi]
        // B-layout state:   bst[kk][v] = G[k=4kk+v+2hi, n=lo] = h0[b0+lo, k]
        v2f wm[4], bst[4];
        #pragma unroll
        for (int kk = 0; kk < 4; ++kk) {
            const int k = 4 * kk + 2 * hi;
            v2f w = { W_hh[lo * NH + k], W_hh[lo * NH + k + 1] };
            v2f g = { h0[(b0 + lo) * NH + k], h0[(b0 + lo) * NH + k + 1] };
            wm[kk]  = w;
            bst[kk] = g;
        }
        const v8f vz = {};

        for (int c = 0; c < NCH; ++c) {
            const int bufi = c & 1;
            float* hb = hist[bufi];
            // preload chunk's x row, pre-masked for the B5 operand
            const float4* xrow = (const float4*)(&xbuf[bufi][lo * CH]);
            float xr[CH];
            #pragma unroll
            for (int q = 0; q < 4; ++q) {
                const float4 t = xrow[q];
                xr[4 * q + 0] = hi ? 0.0f : t.x;
                xr[4 * q + 1] = hi ? 0.0f : t.y;
                xr[4 * q + 2] = hi ? 0.0f : t.z;
                xr[4 * q + 3] = hi ? 0.0f : t.w;
            }
            #pragma unroll
            for (int tt = 0; tt < CH; ++tt) {
                v2f b5 = { xr[tt], oneMask };     // B5 = [x_t ; 1 ; 0 ; 0]
                // xp wmma: no dependence on bst -> issues ahead of the chain
                v8f c0 = __builtin_amdgcn_wmma_f32_16x16x4_f32(
                            false, a5, false, b5, (short)0, vz, false, false);
                v8f c1 = __builtin_amdgcn_wmma_f32_16x16x4_f32(
                            false, wm[2], false, bst[2], (short)0, vz, false, false);
                c0 = __builtin_amdgcn_wmma_f32_16x16x4_f32(
                            false, wm[0], false, bst[0], (short)0, c0, false, false);
                c1 = __builtin_amdgcn_wmma_f32_16x16x4_f32(
                            false, wm[3], false, bst[3], (short)0, c1, false, false);
                c0 = __builtin_amdgcn_wmma_f32_16x16x4_f32(
                            false, wm[1], false, bst[1], (short)0, c0, false, false);
                float d[8];
                #pragma unroll
                for (int r = 0; r < 8; ++r)
                    d[r] = fast_tanh(c0[r] + c1[r]);
                // history store, batch-major: row b=lo gets j = 8*hi + [0..7]
                // two contiguous float4 stores, 16B aligned
                float* sp = hb + tt * SLOT + lo * RSTR + 8 * hi;
                float4 s0; s0.x = d[0]; s0.y = d[1]; s0.z = d[2]; s0.w = d[3];
                float4 s1; s1.x = d[4]; s1.y = d[5]; s1.z = d[6]; s1.w = d[7];
                ((float4*)sp)[0] = s0;
                ((float4*)sp)[1] = s1;
                // D-layout -> B-layout: 4 lane-half swaps produce all 8 regs
                half_swap_pair(d[0], d[2], hi);   // -> b[0][0], b[2][0]
                half_swap_pair(d[1], d[3], hi);   // -> b[0][1], b[2][1]
                half_swap_pair(d[4], d[6], hi);   // -> b[1][0], b[3][0]
                half_swap_pair(d[5], d[7], hi);   // -> b[1][1], b[3][1]
                v2f n0 = { d[0], d[1] };  bst[0] = n0;
                v2f n2 = { d[2], d[3] };  bst[2] = n2;
                v2f n1 = { d[4], d[5] };  bst[1] = n1;
                v2f n3 = { d[6], d[7] };  bst[3] = n3;
            }
            __syncthreads();
        }
        // final hidden state h_T: bst[kk][v] = G_T[k=4kk+v+2hi, b=lo]
        #pragma unroll
        for (int kk = 0; kk < 4; ++kk) {
            const int k = 4 * kk + 2 * hi;
            out[(size_t)NB * NT + (size_t)(b0 + lo) * NH + k]     = bst[kk][0];
            out[(size_t)NB * NT + (size_t)(b0 + lo) * NH + k + 1] = bst[kk][1];
        }
    } else {
        float wout[NH];
        #pragma unroll
        for (int n = 0; n < NH; ++n) wout[n] = W_out[n];
        const float bo = b_out[0];

        for (int c = 0; c < NCH; ++c) {
            if (c + 1 < NCH) {   // prefetch next x chunk
                #pragma unroll
                for (int i = 0; i < 2; ++i) {
                    const int idx = lane + 32 * i;
                    const int m = idx >> 2, q = idx & 3;
                    const float4 v = ((const float4*)(x + (size_t)(b0 + m) * NT
                                                        + (size_t)(c + 1) * CH))[q];
                    ((float4*)(&xbuf[(c + 1) & 1][m * CH]))[q] = v;
                }
            }
            if (c >= 1) {        // project previous chunk's history -> out
                const float* hb = hist[(c - 1) & 1];
                const int t0p = (c - 1) * CH;
                #pragma unroll
                for (int j8 = 0; j8 < 8; ++j8) {
                    const int o = lane + 32 * j8;
                    const int m = o >> 4, tt = o & 15;
                    const float4* row = (const float4*)(hb + tt * SLOT + m * RSTR);
                    const float4 r0 = row[0], r1 = row[1], r2 = row[2], r3 = row[3];
                    float s = bo;
                    s = __builtin_fmaf(r0.x, wout[0],  s);
                    s = __builtin_fmaf(r0.y, wout[1],  s);
                    s = __builtin_fmaf(r0.z, wout[2],  s);
                    s = __builtin_fmaf(r0.w, wout[3],  s);
                    s = __builtin_fmaf(r1.x, wout[4],  s);
                    s = __builtin_fmaf(r1.y, wout[5],  s);
                    s = __builtin_fmaf(r1.z, wout[6],  s);
                    s = __builtin_fmaf(r1.w, wout[7],  s);
                    s = __builtin_fmaf(r2.x, wout[8],  s);
                    s = __builtin_fmaf(r2.y, wout[9],  s);
                    s = __builtin_fmaf(r2.z, wout[10], s);
                    s = __builtin_fmaf(r2.w, wout[11], s);
                    s = __builtin_fmaf(r3.x, wout[12], s);
                    s = __builtin_fmaf(r3.y, wout[13], s);
                    s = __builtin_fmaf(r3.z, wout[14], s);
                    s = __builtin_fmaf(r3.w, wout[15], s);
                    out[(size_t)(b0 + m) * NT + t0p + tt] = s;
                }
            }
            __syncthreads();
        }
        // epilogue: project the last chunk
        const float* hb = hist[(NCH - 1) & 1];
        const int t0p = (NCH - 1) * CH;
        #pragma unroll
        for (int j8 = 0; j8 < 8; ++j8) {
            const int o = lane + 32 * j8;
            const int m = o >> 4, tt = o & 15;
            const float4* row = (const float4*)(hb + tt * SLOT + m * RSTR);
            const float4 r0 = row[0], r1 = row[1], r2 = row[2], r3 = row[3];
            float s = bo;
            s = __builtin_fmaf(r0.x, wout[0],  s);
            s = __builtin_fmaf(r0.y, wout[1],  s);
            s = __builtin_fmaf(r0.z, wout[2],  s);
            s = __builtin_fmaf(r0.w, wout[3],  s);
            s = __builtin_fmaf(r1.x, wout[4],  s);
            s = __builtin_fmaf(r1.y, wout[5],  s);
            s = __builtin_fmaf(r1.z, wout[6],  s);
            s = __builtin_fmaf(r1.w, wout[7],  s);
            s = __builtin_fmaf(r2.x, wout[8],  s);
            s = __builtin_fmaf(r2.y, wout[9],  s);
            s = __builtin_fmaf(r2.z, wout[10], s);
            s = __builtin_fmaf(r2.w, wout[11], s);
            s = __builtin_fmaf(r3.x, wout[12], s);
            s = __builtin_fmaf(r3.y, wout[13], s);
            s = __builtin_fmaf(r3.z, wout[14], s);
            s = __builtin_fmaf(r3.w, wout[15], s);
            out[(size_t)(b0 + m) * NT + t0p + tt] = s;
        }
    }
}

extern "C" void kernel_launch(void* const* d_in, const int* in_sizes, int n_in,
                              void* d_out, int out_size, void* d_ws, size_t ws_size,
                              hipStream_t stream) {
    (void)in_sizes; (void)n_in; (void)out_size; (void)d_ws; (void)ws_size;
    const float* x     = (const float*)d_in[0];
    const float* h0    = (const float*)d_in[1];
    const float* W_ih  = (const float*)d_in[2];
    const float* W_hh  = (const float*)d_in[3];
    const float* b_ih  = (const float*)d_in[4];
    const float* b_hh  = (const float*)d_in[5];
    const float* W_out = (const float*)d_in[6];
    const float* b_out = (const float*)d_in[7];
    float* out = (float*)d_out;

    rnn_fused_wmma<<<dim3(NB / 16), dim3(64), 0, stream>>>(
        x, h0, W_ih, W_hh, b_ih, b_hh, W_out, b_out, out);
}